// Retarget_12996571038226
// MI455X (gfx1250) — compile-verified
//
#include <hip/hip_runtime.h>
#include <math.h>

// CDNA5 / gfx1250. wave32 only. Uses V_WMMA_F32_16X16X4_F32 for the three
// 64x64 = (64x16)x(16x16)x(16x64) factorized Gaussian-conv GEMMs.

typedef float v2f __attribute__((ext_vector_type(2)));
typedef float v8f __attribute__((ext_vector_type(8)));

#define NB    16     // batch
#define HS    16     // saliency H=W
#define HW    64     // feature-map H=W (= grid_size)
#define CH    256    // channels
#define GPAD  3

// ---------------------------------------------------------------------------
// Setup: build the two constant 64x16 operators
//   M  = Conv7 * ReflectPad * BilinearUp   (p_filter path)
//   Mx = Conv7 * diag(coords) * ReflectPad * BilinearUp   (coordinate path)
// jax.image.resize(linear): sample_f = (i+0.5)/4 - 0.5, triangle kernel,
// column-normalized. Reflect pad: t<0 -> -t ; t>63 -> 126-t.
// ---------------------------------------------------------------------------
__global__ void setup_mats(float* __restrict__ M, float* __restrict__ Mx) {
    const float ln2x4 = 2.772588722239781f; // 4*ln(2)
    for (int e = threadIdx.x; e < 64 * 16; e += blockDim.x) {
        int i = e >> 4;   // output row 0..63
        int k = e & 15;   // input col 0..15
        float accM = 0.f, accMx = 0.f;
        for (int t = 0; t < 7; ++t) {
            int j = i + t;                         // padded index 0..69
            float coord = (float)(j - GPAD) / 63.0f;
            int tt = j - GPAD;
            int r = (tt < 0) ? -tt : ((tt > 63) ? 126 - tt : tt);
            float sf = ((float)r + 0.5f) * 0.25f - 0.5f;
            float wnum = fmaxf(0.f, 1.f - fabsf(sf - (float)k));
            float wden = 0.f;
            for (int kp = 0; kp < 16; ++kp)
                wden += fmaxf(0.f, 1.f - fabsf(sf - (float)kp));
            float u = wnum / wden;
            float dt = (float)(t - GPAD);
            float g = expf(-ln2x4 * dt * dt / 169.0f);  // FWHM=13 -> 169
            accM  += g * u;
            accMx += g * coord * u;
        }
        M[e]  = accM;
        Mx[e] = accMx;
    }
}

// ---------------------------------------------------------------------------
// Main kernel: one wave32 per (b,c) pair. 4 waves / 128 threads per block.
// ---------------------------------------------------------------------------
#define WAVES 4

static __device__ __forceinline__ v8f wmma4(v2f a, v2f b, v8f c) {
    return __builtin_amdgcn_wmma_f32_16x16x4_f32(
        /*neg_a=*/false, a, /*neg_b=*/false, b,
        /*c_mod=*/(short)0, c, /*reuse_a=*/false, /*reuse_b=*/false);
}

__global__ __launch_bounds__(32 * WAVES)
void retarget_main(const float* __restrict__ fm, const float* __restrict__ sal,
                   float* __restrict__ out,
                   const float* __restrict__ Mg, const float* __restrict__ Mxg) {
    __shared__ float Msm [64][17];
    __shared__ float Mxsm[64][17];
    __shared__ float Ssm [WAVES][16][16];
    __shared__ float T1sm[WAVES][64][17];
    __shared__ float T2sm[WAVES][64][17];

    const int tid  = threadIdx.x;
    const int wv   = tid >> 5;
    const int lane = tid & 31;
    const int row  = lane & 15;   // M-row for A frags, N-col for B/D frags
    const int hi   = lane >> 4;   // half-wave selector (K+2 / M+8 halves)

    // Block-cooperative load of the constant operators into padded LDS.
    for (int e = tid; e < 1024; e += 32 * WAVES) {
        Msm [e >> 4][e & 15] = Mg[e];
        Mxsm[e >> 4][e & 15] = Mxg[e];
    }

    const int pair = blockIdx.x * WAVES + wv;   // grid sized exactly: 4096 pairs
    const int b = pair >> 8;                     // pair / 256
    const int c = pair & 255;                    // pair % 256

    // Per-wave load of the 16x16 saliency slice (+1e-18 as in reference).
    for (int e = lane; e < 256; e += 32) {
        int r = e >> 4, col = e & 15;
        Ssm[wv][r][col] = sal[((b * HS + r) * HS + col) * CH + c] + 1e-18f;
    }
    __syncthreads();

    // ---- Stage 1: T1 = M*S , T2 = Mx*S  (64x16 each), K=16 in 4 WMMA steps.
    for (int ri = 0; ri < 4; ++ri) {
        v8f acc1 = {};
        v8f acc2 = {};
#pragma unroll
        for (int kk = 0; kk < 16; kk += 4) {
            const int kc = kk + 2 * hi;
            v2f a1, a2, bb;
            a1.x = Msm [ri * 16 + row][kc];  a1.y = Msm [ri * 16 + row][kc + 1];
            a2.x = Mxsm[ri * 16 + row][kc];  a2.y = Mxsm[ri * 16 + row][kc + 1];
            bb.x = Ssm[wv][kc][row];         bb.y = Ssm[wv][kc + 1][row];
            acc1 = wmma4(a1, bb, acc1);
            acc2 = wmma4(a2, bb, acc2);
        }
#pragma unroll
        for (int j = 0; j < 8; ++j) {
            T1sm[wv][ri * 16 + j + 8 * hi][row] = acc1[j];
            T2sm[wv][ri * 16 + j + 8 * hi][row] = acc2[j];
        }
    }
    __syncthreads();

    // ---- Stage 2: per 16x16 output tile (ri,cj):
    //   p_filter = T1 * M^T ; x_mul_x = T1 * Mx^T ; x_mul_y = T2 * M^T
    // fused with grid math + gather + store (weight == (x1-x0)*(y1-y0)).
    const long fbase = (long)b * (HW * HW * CH);
    for (int ri = 0; ri < 4; ++ri) {
        for (int cj = 0; cj < 4; ++cj) {
            v8f pf = {};
            v8f xm = {};
            v8f ym = {};
#pragma unroll
            for (int kk = 0; kk < 16; kk += 4) {
                const int kc = kk + 2 * hi;
                v2f a1, a2, bM, bMx;
                a1.x  = T1sm[wv][ri * 16 + row][kc];  a1.y  = T1sm[wv][ri * 16 + row][kc + 1];
                a2.x  = T2sm[wv][ri * 16 + row][kc];  a2.y  = T2sm[wv][ri * 16 + row][kc + 1];
                // B[k][n] = (M^T)[k][cj*16+n] = M[cj*16+n][k]
                bM.x  = Msm [cj * 16 + row][kc];      bM.y  = Msm [cj * 16 + row][kc + 1];
                bMx.x = Mxsm[cj * 16 + row][kc];      bMx.y = Mxsm[cj * 16 + row][kc + 1];
                pf = wmma4(a1, bM,  pf);
                xm = wmma4(a1, bMx, xm);
                ym = wmma4(a2, bM,  ym);
            }
#pragma unroll
            for (int j = 0; j < 8; ++j) {
                float P = pf[j], X = xm[j], Y = ym[j];
                // safe_div: den==0 -> 0, then 2*0-1 = -1 (inside clip range)
                float xg = (P != 0.f) ? (2.f * X / P - 1.f) : -1.f;
                float yg = (P != 0.f) ? (2.f * Y / P - 1.f) : -1.f;
                xg = fminf(1.f, fmaxf(-1.f, xg));
                yg = fminf(1.f, fmaxf(-1.f, yg));
                float xf = 0.5f * (xg + 1.f) * 62.f;   // (max_x - 1) = 62
                float yf = 0.5f * (yg + 1.f) * 62.f;
                int x0 = (int)floorf(xf + 0.5f);
                int y0 = (int)floorf(yf + 0.5f);
                int x1 = min(max(x0 + 1, 0), 63);
                int y1 = min(max(y0 + 1, 0), 63);
                x0 = min(max(x0, 0), 63);
                y0 = min(max(y0, 0), 63);
                float w = (float)((x1 - x0) * (y1 - y0));
                int yy = ri * 16 + j + 8 * hi;
                int xx = cj * 16 + row;
                float v = fm[fbase + (long)(y0 * HW + x0) * CH + c];
                out[fbase + (long)(yy * HW + xx) * CH + c] = w * v;
            }
        }
    }
}

// ---------------------------------------------------------------------------
extern "C" void kernel_launch(void* const* d_in, const int* in_sizes, int n_in,
                              void* d_out, int out_size, void* d_ws, size_t ws_size,
                              hipStream_t stream) {
    const float* fm  = (const float*)d_in[0];  // (16,64,64,256) f32
    const float* sal = (const float*)d_in[1];  // (16,16,16,256) f32
    float* out = (float*)d_out;                // (16,64,64,256) f32

    float* Mws  = (float*)d_ws;                // 64*16 floats
    float* Mxws = Mws + 64 * 16;               // 64*16 floats

    setup_mats<<<1, 256, 0, stream>>>(Mws, Mxws);

    const int pairs  = NB * CH;                // 4096
    const int blocks = pairs / WAVES;          // 1024
    retarget_main<<<blocks, 32 * WAVES, 0, stream>>>(fm, sal, out, Mws, Mxws);
}